// Head_43258910605999
// MI455X (gfx1250) — compile-verified
//
#include <hip/hip_runtime.h>

typedef __attribute__((ext_vector_type(2))) float v2f;
typedef __attribute__((ext_vector_type(8))) float v8f;

constexpr int T_DIM = 256;   // sequence length
constexpr int C_DIM = 384;   // input channels
constexpr int H_DIM = 64;    // head size
constexpr int LDW   = 68;    // padded LDS row stride (floats) for Q/K/V (68 % 64 == 4 -> conflict-free column reads)
constexpr int PLD   = 17;    // P staging tile row stride (17 coprime with 64)

__device__ __forceinline__ v8f wmma_f32(v2f a, v2f b, v8f c) {
    return __builtin_amdgcn_wmma_f32_16x16x4_f32(false, a, false, b, (short)0, c, false, false);
}

__global__ __launch_bounds__(512, 1)
void attn_head_fused(const float* __restrict__ x,
                     const float* __restrict__ Wq,
                     const float* __restrict__ Wk,
                     const float* __restrict__ Wv,
                     float* __restrict__ out)
{
    extern __shared__ float smem[];
    float* Qs = smem;                    // [256][68]
    float* Ks = Qs + T_DIM * LDW;        // [256][68]
    float* Vs = Ks + T_DIM * LDW;        // [256][68]
    float* Ps = Vs + T_DIM * LDW;        // 16 waves x [16][17] staging

    const int b    = blockIdx.x;
    const int tid  = threadIdx.x;
    const int w    = tid >> 5;           // wave id, 0..15; owns rows [16w, 16w+16)
    const int lane = tid & 31;
    const int l16  = lane & 15;
    const int hi   = lane >> 4;          // half-wave select
    const int kofs = hi << 1;            // K offset within group of 4 (A/B frag layout)
    const int r0   = w << 4;             // row-tile base

    const float* xb = x + (size_t)b * T_DIM * C_DIM;

    // ---------------- Stage 1: Q,K,V = x @ W (per-wave 16-row tile) ----------------
    v8f acc[3][4];
    #pragma unroll
    for (int m = 0; m < 3; ++m)
        #pragma unroll
        for (int n = 0; n < 4; ++n)
            acc[m][n] = (v8f)0.0f;

    const float* Wm[3] = { Wq, Wk, Wv };
    const float* arow  = xb + (size_t)(r0 + l16) * C_DIM;

    for (int kc = 0; kc < C_DIM / 4; ++kc) {
        const int kb = kc * 4 + kofs;
        v2f a;
        a.x = arow[kb];
        a.y = arow[kb + 1];
        #pragma unroll
        for (int m = 0; m < 3; ++m) {
            const float* Wp = Wm[m] + (size_t)kb * H_DIM + l16;
            #pragma unroll
            for (int n = 0; n < 4; ++n) {
                v2f bb;
                bb.x = Wp[n * 16];
                bb.y = Wp[H_DIM + n * 16];
                acc[m][n] = wmma_f32(a, bb, acc[m][n]);
            }
        }
    }

    // Spill D-layout accumulators to padded LDS (row-major)
    {
        float* dst[3] = { Qs, Ks, Vs };
        #pragma unroll
        for (int m = 0; m < 3; ++m)
            #pragma unroll
            for (int n = 0; n < 4; ++n)
                #pragma unroll
                for (int r = 0; r < 8; ++r) {
                    const int row = r0 + r + (hi << 3);
                    const int col = n * 16 + l16;
                    dst[m][row * LDW + col] = acc[m][n][r];
                }
    }
    __syncthreads();

    // ---------------- Stage 2: causal attention for rows [r0, r0+16) ----------------
    const float scale = 19.595917942265423f;     // C ** 0.5 (source multiplies!)
    const int   jmax  = w;                        // causal column-tile bound

    v8f sacc[16];
    #pragma unroll
    for (int j = 0; j < 16; ++j) {
        if (j <= jmax) {
            v8f s = (v8f)0.0f;
            #pragma unroll
            for (int kc = 0; kc < H_DIM / 4; ++kc) {
                const int kb = kc * 4 + kofs;
                v2f a, bb;
                a.x  = Qs[(r0 + l16) * LDW + kb];
                a.y  = Qs[(r0 + l16) * LDW + kb + 1];
                bb.x = Ks[(j * 16 + l16) * LDW + kb];   // B = K^T
                bb.y = Ks[(j * 16 + l16) * LDW + kb + 1];
                s = wmma_f32(a, bb, s);
            }
            #pragma unroll
            for (int r = 0; r < 8; ++r) {
                float v = s[r] * scale;
                if (j == jmax) {
                    const int row = r0 + r + (hi << 3);
                    const int col = j * 16 + l16;
                    if (col > row) v = -1e30f;
                }
                s[r] = v;
            }
            sacc[j] = s;
        }
    }

    // Row softmax: reductions over register strip + 16-lane half-wave shuffles
    float mx[8], sm[8];
    #pragma unroll
    for (int r = 0; r < 8; ++r) mx[r] = -3.0e38f;
    #pragma unroll
    for (int j = 0; j < 16; ++j)
        if (j <= jmax)
            #pragma unroll
            for (int r = 0; r < 8; ++r) mx[r] = fmaxf(mx[r], sacc[j][r]);
    #pragma unroll
    for (int r = 0; r < 8; ++r) {
        float m = mx[r];
        #pragma unroll
        for (int off = 1; off < 16; off <<= 1)
            m = fmaxf(m, __shfl_xor(m, off, 32));
        mx[r] = m;
        sm[r] = 0.0f;
    }
    #pragma unroll
    for (int j = 0; j < 16; ++j)
        if (j <= jmax)
            #pragma unroll
            for (int r = 0; r < 8; ++r) {
                const float e = __expf(sacc[j][r] - mx[r]);
                sacc[j][r] = e;
                sm[r] += e;
            }
    #pragma unroll
    for (int r = 0; r < 8; ++r) {
        float s2 = sm[r];
        #pragma unroll
        for (int off = 1; off < 16; off <<= 1)
            s2 += __shfl_xor(s2, off, 32);
        sm[r] = s2;
    }

    // O = P @ V  (P tiles: D-layout -> A-fragment via per-wave LDS staging)
    v8f oacc[4];
    #pragma unroll
    for (int n = 0; n < 4; ++n) oacc[n] = (v8f)0.0f;

    float* Pw = Ps + w * 16 * PLD;
    #pragma unroll
    for (int j = 0; j < 16; ++j) {
        if (j <= jmax) {
            #pragma unroll
            for (int r = 0; r < 8; ++r)
                Pw[(r + (hi << 3)) * PLD + l16] = sacc[j][r];
            // LDS ops are wave-in-order; keep compiler from reordering the
            // cross-lane store->load and drain DScnt explicitly.
            asm volatile("s_wait_dscnt 0x0" ::: "memory");
            #pragma unroll
            for (int kc = 0; kc < 4; ++kc) {
                const int kb = kc * 4 + kofs;
                v2f a;
                a.x = Pw[l16 * PLD + kb];
                a.y = Pw[l16 * PLD + kb + 1];
                #pragma unroll
                for (int nt = 0; nt < 4; ++nt) {
                    v2f bb;
                    bb.x = Vs[(j * 16 + kb) * LDW + nt * 16 + l16];
                    bb.y = Vs[(j * 16 + kb + 1) * LDW + nt * 16 + l16];
                    oacc[nt] = wmma_f32(a, bb, oacc[nt]);
                }
            }
        }
    }

    // Normalize by row-sum and store [B,T,H]
    float* ob = out + (size_t)b * T_DIM * H_DIM;
    float rinv[8];
    #pragma unroll
    for (int r = 0; r < 8; ++r) rinv[r] = 1.0f / sm[r];
    #pragma unroll
    for (int nt = 0; nt < 4; ++nt)
        #pragma unroll
        for (int r = 0; r < 8; ++r) {
            const int row = r0 + r + (hi << 3);
            ob[(size_t)row * H_DIM + nt * 16 + l16] = oacc[nt][r] * rinv[r];
        }
}

extern "C" void kernel_launch(void* const* d_in, const int* in_sizes, int n_in,
                              void* d_out, int out_size, void* d_ws, size_t ws_size,
                              hipStream_t stream) {
    const float* x  = (const float*)d_in[0];
    const float* Wq = (const float*)d_in[1];
    const float* Wk = (const float*)d_in[2];
    const float* Wv = (const float*)d_in[3];
    float* out = (float*)d_out;

    const int B = 512;
    const size_t smem_bytes = (size_t)(3 * T_DIM * LDW + 16 * 16 * PLD) * sizeof(float); // ~226 KB < 320 KB/WGP
    hipLaunchKernelGGL(attn_head_fused, dim3(B), dim3(512), smem_bytes, stream,
                       x, Wq, Wk, Wv, out);
}